// SpatioTemporalGCN_88270167867686
// MI455X (gfx1250) — compile-verified
//
#include <hip/hip_runtime.h>
#include <hip/hip_bf16.h>

typedef float v2f __attribute__((ext_vector_type(2)));
typedef float v8f __attribute__((ext_vector_type(8)));

#define NNODES 5000
#define BATCH  16
#define LAGN   12
#define DINC   32
#define EDIM   64
#define HHC    32
#define HOUT   64
#define LN_EPS 1e-5f

// LDS row strides (floats) chosen for bank-conflict-free fragment reads
#define AS_STRIDE 12   // A panel: 128 rows x 8 k, stride 12
#define BS_STRIDE 72   // B panel: 8 k x 64 cols, stride 72

// ---------------------------------------------------------------- utilities
__device__ __forceinline__ float wave_sum32(float v) {
    #pragma unroll
    for (int m = 16; m > 0; m >>= 1) v += __shfl_xor(v, m, 32);
    return v;
}

// ---------------------------------------------------------------- K0: E^T
__global__ void k_transpose_E(const float* __restrict__ E, float* __restrict__ ET) {
    int idx = blockIdx.x * blockDim.x + threadIdx.x;        // over EDIM*NNODES
    if (idx >= EDIM * NNODES) return;
    int d = idx / NNODES, n = idx % NNODES;
    ET[idx] = E[n * EDIM + d];
}

// ---------------------------------------------------------------- K1: pack x -> Xt[n, b*32+c]
__global__ void k_pack_x(const float* __restrict__ x, float* __restrict__ Xt) {
    int idx = blockIdx.x * blockDim.x + threadIdx.x;        // over NNODES*512
    if (idx >= NNODES * BATCH * DINC) return;
    int n = idx / (BATCH * DINC);
    int r = idx % (BATCH * DINC);
    int b = r / DINC, c = r % DINC;
    Xt[idx] = x[((size_t)b * NNODES + n) * DINC + c];
}

// ---------------------------------------------------------------- K2: collapse LAG with T
__global__ void k_window(const float* __restrict__ xw, const float* __restrict__ T,
                         float* __restrict__ XW) {
    int idx = blockIdx.x * blockDim.x + threadIdx.x;        // over B*N*16
    if (idx >= BATCH * NNODES * 16) return;
    int i = idx & 15;
    int rest = idx >> 4;
    int n = rest % NNODES;
    int b = rest / NNODES;
    float acc = 0.f;
    #pragma unroll
    for (int t = 0; t < LAGN; ++t)
        acc += T[t] * xw[(((size_t)b * LAGN + t) * NNODES + n) * 16 + i];
    XW[idx] = acc;
}

// ---------------------------------------------------------------- K3: WMMA f32 GEMM (LDS staged, pipelined)
// C[M,N] = rowScale(row) * (A[M,K] @ B[K,N]), optional relu. Requires K % 8 == 0.
// Block: 128 threads = 4 waves. Block tile 128x64; wave tile 32x64 (two 16-row
// fragment groups sharing each B fragment); K chunk 8.
// OOB handled purely by index clamping (no EXEC divergence around WMMA).
__global__ void __launch_bounds__(128)
k_wmma_gemm(const float* __restrict__ A, const float* __restrict__ B,
            float* __restrict__ C, int M, int N, int K,
            int lda, int ldb, int ldc,
            const float* __restrict__ rowScale, int doRelu) {
    __shared__ float As[128 * AS_STRIDE];
    __shared__ float Bs[8 * BS_STRIDE];

    const int t    = threadIdx.x;
    const int lane = t & 31;
    const int wave = t >> 5;                     // 0..3
    const int colBlocks = (N + 63) >> 6;
    const int bm = blockIdx.x / colBlocks;
    const int bn = blockIdx.x % colBlocks;
    const int row0 = bm << 7;                    // 128-row panel
    const int col0 = bn << 6;                    // 64-col panel

    // --- staging indices: A = 2 x float4 per thread, B = 1 x float4 per thread
    const int ar0 = t >> 1;                      // rows 0..63   (idx = t)
    const int ar1 = (t + 128) >> 1;              // rows 64..127 (idx = t+128)
    const int ac0 = (t & 1) << 2;                // 0 or 4
    const int aRowG0 = min(row0 + ar0, M - 1);   // clamp; garbage rows never stored
    const int aRowG1 = min(row0 + ar1, M - 1);
    const int br = t >> 4;                       // 0..7
    const int bc = (t & 15) << 2;                // 0..60
    const int bColG = min(col0 + bc, N - 4);     // clamp; garbage cols never stored

    // --- fragment indices
    const int kHalf = (lane >> 4) << 1;          // 0 or 2
    const int nSub  = lane & 15;
    const int aBase0 = (wave * 32 + nSub) * AS_STRIDE;        // rows wave*32 + 0..15
    const int aBase1 = (wave * 32 + 16 + nSub) * AS_STRIDE;   // rows wave*32 + 16..31

    v8f acc[2][4];
    #pragma unroll
    for (int g = 0; g < 2; ++g)
        #pragma unroll
        for (int s = 0; s < 4; ++s) acc[g][s] = v8f{0.f,0.f,0.f,0.f,0.f,0.f,0.f,0.f};

    // prefetch chunk 0
    float4 av0 = *(const float4*)(A + (size_t)aRowG0 * lda + ac0);
    float4 av1 = *(const float4*)(A + (size_t)aRowG1 * lda + ac0);
    float4 bv  = *(const float4*)(B + (size_t)br * ldb + bColG);

    for (int k = 0; k < K; k += 8) {
        __syncthreads();                         // previous compute done reading LDS
        *(float4*)&As[ar0 * AS_STRIDE + ac0] = av0;
        *(float4*)&As[ar1 * AS_STRIDE + ac0] = av1;
        *(float4*)&Bs[br  * BS_STRIDE + bc]  = bv;
        if (k + 8 < K) {                         // uniform branch: prefetch next chunk
            av0 = *(const float4*)(A + (size_t)aRowG0 * lda + (k + 8) + ac0);
            av1 = *(const float4*)(A + (size_t)aRowG1 * lda + (k + 8) + ac0);
            bv  = *(const float4*)(B + (size_t)(k + 8 + br) * ldb + bColG);
        }
        __syncthreads();                         // staged data visible
        #pragma unroll
        for (int kc = 0; kc < 8; kc += 4) {
            v2f afrag0 = *(const v2f*)&As[aBase0 + kc + kHalf];
            v2f afrag1 = *(const v2f*)&As[aBase1 + kc + kHalf];
            #pragma unroll
            for (int s = 0; s < 4; ++s) {
                v2f bfrag;
                bfrag.x = Bs[(kc + kHalf)     * BS_STRIDE + s * 16 + nSub];
                bfrag.y = Bs[(kc + kHalf + 1) * BS_STRIDE + s * 16 + nSub];
                acc[0][s] = __builtin_amdgcn_wmma_f32_16x16x4_f32(
                    false, afrag0, false, bfrag, (short)0, acc[0][s], false, false);
                acc[1][s] = __builtin_amdgcn_wmma_f32_16x16x4_f32(
                    false, afrag1, false, bfrag, (short)0, acc[1][s], false, false);
            }
        }
    }

    // --- epilogue: wave w owns rows row0 + w*32 .. +31 (two 16-row groups)
    const int rowHi = (lane >> 4) << 3;
    #pragma unroll
    for (int g = 0; g < 2; ++g) {
        const int wrow0 = row0 + wave * 32 + g * 16;
        #pragma unroll
        for (int s = 0; s < 4; ++s) {
            int col = col0 + s * 16 + nSub;
            if (col >= N) continue;
            #pragma unroll
            for (int j = 0; j < 8; ++j) {
                int row = wrow0 + j + rowHi;
                if (row >= M) continue;
                float v = acc[g][s][j];
                if (rowScale) v *= rowScale[row];
                if (doRelu)   v = v > 0.f ? v : 0.f;
                C[(size_t)row * ldc + col] = v;
            }
        }
    }
}

// ---------------------------------------------------------------- K4: row softmax (exp in place + inv sum)
__global__ void __launch_bounds__(256)
k_row_softmax(float* __restrict__ G, float* __restrict__ invSum) {
    int n = blockIdx.x;
    float* row = G + (size_t)n * NNODES;
    __shared__ float red[256];
    int t = threadIdx.x;
    float m = 0.f;                                   // relu output >= 0
    for (int j = t; j < NNODES; j += 256) m = fmaxf(m, row[j]);
    red[t] = m; __syncthreads();
    for (int s = 128; s > 0; s >>= 1) {
        if (t < s) red[t] = fmaxf(red[t], red[t + s]);
        __syncthreads();
    }
    float rmax = red[0]; __syncthreads();
    float sum = 0.f;
    for (int j = t; j < NNODES; j += 256) {
        float e = expf(row[j] - rmax);
        row[j] = e;
        sum += e;
    }
    red[t] = sum; __syncthreads();
    for (int s = 128; s > 0; s >>= 1) {
        if (t < s) red[t] += red[t + s];
        __syncthreads();
    }
    if (t == 0) invSum[n] = 1.f / red[0];
}

// ---------------------------------------------------------------- K5: fused per-node epilogue
__global__ void __launch_bounds__(256)
k_final(const float* __restrict__ x, const float* __restrict__ y1,
        const float* __restrict__ y2, const float* __restrict__ XW,
        const float* __restrict__ E,  const float* __restrict__ WP,
        const float* __restrict__ WW, const float* __restrict__ BP,
        const float* __restrict__ ln1w, const float* __restrict__ ln1b,
        const float* __restrict__ ln2w, const float* __restrict__ ln2b,
        float* __restrict__ out) {
    __shared__ float eRow[EDIM];
    __shared__ float Wl[96 * HHC];      // 3072 dynamic weights for this node
    __shared__ float WWl[16 * HHC];     // 512 window weights
    __shared__ float biasl[HOUT];

    const int n = blockIdx.x;
    const int t = threadIdx.x;
    if (t < EDIM) eRow[t] = E[(size_t)n * EDIM + t];
    __syncthreads();

    // W_n[j,o] = sum_d E[n,d] * WP[d, j*32+o]  (WP row-major [64,3072], L2 resident)
    {
        float acc[12];
        #pragma unroll
        for (int r = 0; r < 12; ++r) acc[r] = 0.f;
        for (int d = 0; d < EDIM; ++d) {
            float e = eRow[d];
            const float* wp = WP + (size_t)d * 3072;
            #pragma unroll
            for (int r = 0; r < 12; ++r) acc[r] += e * wp[t + r * 256];
        }
        #pragma unroll
        for (int r = 0; r < 12; ++r) Wl[t + r * 256] = acc[r];
    }
    // ww_n[i,o] = sum_d E[n,d] * WW[d, i*32+o]   ([64,512])
    {
        float a0 = 0.f, a1 = 0.f;
        for (int d = 0; d < EDIM; ++d) {
            float e = eRow[d];
            const float* wwp = WW + (size_t)d * 512;
            a0 += e * wwp[t];
            a1 += e * wwp[t + 256];
        }
        WWl[t] = a0; WWl[t + 256] = a1;
    }
    // bias_n[o] = sum_d E[n,d] * BP[d,o]   ([64,64])
    if (t < HOUT) {
        float a = 0.f;
        for (int d = 0; d < EDIM; ++d) a += eRow[d] * BP[(size_t)d * HOUT + t];
        biasl[t] = a;
    }
    __syncthreads();

    const int wave = t >> 5, lane = t & 31;
    const float w1 = ln1w[lane], b1 = ln1b[lane];
    const float w2 = ln2w[lane], b2 = ln2b[lane];

    for (int b = wave; b < BATCH; b += 8) {
        // graph-conv channel: lane = output o
        float g = 0.f;
        const float* xb  = x  + ((size_t)b * NNODES + n) * DINC;
        const float* y1n = y1 + (size_t)n * (BATCH * DINC) + b * DINC;
        const float* y2n = y2 + (size_t)n * (BATCH * DINC) + b * DINC;
        #pragma unroll
        for (int i = 0; i < DINC; ++i) g += xb[i]  * Wl[i * HHC + lane];
        #pragma unroll
        for (int i = 0; i < DINC; ++i) g += y1n[i] * Wl[(DINC + i) * HHC + lane];
        #pragma unroll
        for (int i = 0; i < DINC; ++i) g += y2n[i] * Wl[(2 * DINC + i) * HHC + lane];
        // window channel
        float w = 0.f;
        const float* xwn = XW + ((size_t)b * NNODES + n) * 16;
        #pragma unroll
        for (int i = 0; i < 16; ++i) w += xwn[i] * WWl[i * HHC + lane];

        // LayerNorm over the 32 lanes (one half each)
        float mu1 = wave_sum32(g) * (1.f / 32.f);
        float d1  = g - mu1;
        float v1  = wave_sum32(d1 * d1) * (1.f / 32.f);
        float o1  = d1 * rsqrtf(v1 + LN_EPS) * w1 + b1 + biasl[lane];

        float mu2 = wave_sum32(w) * (1.f / 32.f);
        float d2  = w - mu2;
        float v2  = wave_sum32(d2 * d2) * (1.f / 32.f);
        float o2  = d2 * rsqrtf(v2 + LN_EPS) * w2 + b2 + biasl[HHC + lane];

        float* op = out + ((size_t)b * NNODES + n) * HOUT;
        op[lane]       = o1;
        op[HHC + lane] = o2;
    }
}

// ---------------------------------------------------------------- launch
extern "C" void kernel_launch(void* const* d_in, const int* in_sizes, int n_in,
                              void* d_out, int out_size, void* d_ws, size_t ws_size,
                              hipStream_t stream) {
    const float* x    = (const float*)d_in[0];
    const float* xw   = (const float*)d_in[1];
    const float* E    = (const float*)d_in[2];
    const float* WP   = (const float*)d_in[3];
    const float* WW   = (const float*)d_in[4];
    const float* BP   = (const float*)d_in[5];
    const float* T    = (const float*)d_in[6];
    const float* ln1w = (const float*)d_in[7];
    const float* ln1b = (const float*)d_in[8];
    const float* ln2w = (const float*)d_in[9];
    const float* ln2b = (const float*)d_in[10];
    float* out = (float*)d_out;

    float* ws = (float*)d_ws;
    size_t off = 0;
    float* ET  = ws + off; off += (size_t)EDIM * NNODES;                  // 0.32M
    float* Xt  = ws + off; off += (size_t)NNODES * BATCH * DINC;          // 2.56M
    float* G   = ws + off; off += (size_t)NNODES * NNODES;                // 25.0M
    float* y1  = ws + off; off += (size_t)NNODES * BATCH * DINC;          // 2.56M
    float* y2  = ws + off; off += (size_t)NNODES * BATCH * DINC;          // 2.56M
    float* XW  = ws + off; off += (size_t)BATCH * NNODES * 16;            // 1.28M
    float* inv = ws + off; off += NNODES;

    auto blocks = [](size_t total, int bs) { return (unsigned)((total + bs - 1) / bs); };

    k_transpose_E<<<blocks((size_t)EDIM * NNODES, 256), 256, 0, stream>>>(E, ET);
    k_pack_x<<<blocks((size_t)NNODES * BATCH * DINC, 256), 256, 0, stream>>>(x, Xt);
    k_window<<<blocks((size_t)BATCH * NNODES * 16, 256), 256, 0, stream>>>(xw, T, XW);

    auto gemm = [&](const float* A, const float* B, float* C,
                    int M, int N, int K, int lda, int ldb, int ldc,
                    const float* scale, int relu) {
        int rb = (M + 127) / 128, cb = (N + 63) / 64;
        unsigned nb = (unsigned)(rb * cb);
        k_wmma_gemm<<<nb, 128, 0, stream>>>(A, B, C, M, N, K, lda, ldb, ldc, scale, relu);
    };

    // G = relu(E @ E^T)            [5000 x 5000], K = 64
    gemm(E, ET, G, NNODES, NNODES, EDIM, EDIM, NNODES, NNODES, nullptr, 1);
    // exp in place + inverse row sums
    k_row_softmax<<<NNODES, 256, 0, stream>>>(G, inv);
    // y1 = softmax(G) @ Xt         [5000 x 512], K = 5000
    gemm(G, Xt, y1, NNODES, BATCH * DINC, NNODES, NNODES, BATCH * DINC, BATCH * DINC, inv, 0);
    // y2 = softmax(G) @ y1
    gemm(G, y1, y2, NNODES, BATCH * DINC, NNODES, NNODES, BATCH * DINC, BATCH * DINC, inv, 0);

    // fused per-node dynamic weights + gconv/window conv + LN + bias
    k_final<<<NNODES, 256, 0, stream>>>(x, y1, y2, XW, E, WP, WW, BP,
                                        ln1w, ln1b, ln2w, ln2b, out);
}